// GMMGradient_52493090292207
// MI455X (gfx1250) — compile-verified
//
#include <hip/hip_runtime.h>
#include <math.h>

// ---------------------------------------------------------------------------
// GMM density + analytic gradient, B=8192, K=32, C=128, D=2.
//
// Reformulated as two f32 WMMA GEMMs around an elementwise exp:
//   phase 1:  logcomp[b,c] = f(x_b) . M[:,c]      f = [x0^2,x0x1,x1^2,x0,x1,1,0,0]
//   phase 2:  S[b,j]       = comp[b,c] . G[c,j]   (j: P00,P01,P10,P11,Pmu0,Pmu1,pdf)
// using V_WMMA_F32_16X16X4_F32 (fp32 matrix pipe, full precision).
//
// VGPR layouts (CDNA5 ISA 7.12.2):
//   A 16x4 f32  : 2 VGPRs; lanes 0-15 -> M=lane, K={0,1}; lanes 16-31 -> K={2,3}
//   B 4x16 f32  : 2 VGPRs; assumed to mirror the C/D half-split:
//                 VGPR0: lanes0-15 K=0, lanes16-31 K=2; VGPR1: K=1 / K=3
//   C/D 16x16   : 8 VGPRs; VGPR v: lanes0-15 row v, lanes16-31 row v+8
// ---------------------------------------------------------------------------

typedef float v2f __attribute__((ext_vector_type(2)));
typedef float v8f __attribute__((ext_vector_type(8)));

#define GMM_K 32
#define GMM_C 128
#define LOG2PI_F 1.8377866f /* overwritten below by precise value */
static constexpr float kLog2Pi  = 1.8378770664093453f;
static constexpr float kEps     = 5500.0f;
static constexpr float kInvLmb  = 1.0f / 1100.0f;

// ---------------------------------------------------------------------------
// Precompute per-(k,c) GEMM operand matrices from weights/means/covariances.
//   M : [K][8][C]   (phase-1 B operand, feature-major, c contiguous)
//   G : [K][C][16]  (phase-2 B operand, 7 used columns padded to 16)
// ---------------------------------------------------------------------------
__global__ __launch_bounds__(256)
void gmm_precompute(const float* __restrict__ w,
                    const float* __restrict__ mu,
                    const float* __restrict__ cov,
                    float* __restrict__ M,
                    float* __restrict__ G)
{
    int idx = blockIdx.x * blockDim.x + threadIdx.x;   // idx = k*C + c
    if (idx >= GMM_K * GMM_C) return;
    int k = idx >> 7;
    int c = idx & (GMM_C - 1);

    const float* cv = cov + (size_t)idx * 4;
    float c00 = cv[0], c01 = cv[1], c10 = cv[2], c11 = cv[3];
    float det  = c00 * c11 - c01 * c10;          // SPD -> det > 0
    float inv  = 1.0f / det;
    float P00  = c11 * inv;
    float P01  = -c01 * inv;
    float P10  = -c10 * inv;
    float P11  = c00 * inv;
    float m0   = mu[(size_t)idx * 2 + 0];
    float m1   = mu[(size_t)idx * 2 + 1];
    float s    = P01 + P10;
    float logdet = logf(det);
    float lw     = logf(w[idx]);
    float mPm    = P00 * m0 * m0 + s * m0 * m1 + P11 * m1 * m1;

    float* Mk = M + (size_t)k * 8 * GMM_C;
    Mk[0 * GMM_C + c] = -0.5f * P00;
    Mk[1 * GMM_C + c] = -0.5f * s;
    Mk[2 * GMM_C + c] = -0.5f * P11;
    Mk[3 * GMM_C + c] = P00 * m0 + 0.5f * s * m1;
    Mk[4 * GMM_C + c] = 0.5f * s * m0 + P11 * m1;
    Mk[5 * GMM_C + c] = lw - 0.5f * (2.0f * kLog2Pi + logdet) - 0.5f * mPm;
    Mk[6 * GMM_C + c] = 0.0f;
    Mk[7 * GMM_C + c] = 0.0f;

    float* Gc = G + (size_t)idx * 16;
    Gc[0] = P00;
    Gc[1] = P01;
    Gc[2] = P10;
    Gc[3] = P11;
    Gc[4] = P00 * m0 + P01 * m1;
    Gc[5] = P10 * m0 + P11 * m1;
    Gc[6] = 1.0f;                                 // pdf accumulator column
#pragma unroll
    for (int j = 7; j < 16; ++j) Gc[j] = 0.0f;
}

// ---------------------------------------------------------------------------
// Main kernel: block = 128 threads (4 waves). Each wave owns 16 b-rows of one
// keypoint k. grid = (B/64, K).
// ---------------------------------------------------------------------------
__global__ __launch_bounds__(128)
void gmm_wmma(const float* __restrict__ kps,
              const float* __restrict__ M,
              const float* __restrict__ G,
              float* __restrict__ out,
              int Btot)
{
    __shared__ float Gs[GMM_C * 16];          // 8 KB, shared by the block
    __shared__ float compS[4][16 * GMM_C];    // 32 KB, per-wave comp tile
    __shared__ float Ssh[4][16 * 16];         // 4 KB, per-wave result tile

    const int k    = blockIdx.y;
    const int wave = threadIdx.x >> 5;
    const int lane = threadIdx.x & 31;
    const int n    = lane & 15;               // column / row-in-half index
    const int half = lane >> 4;               // 0: lanes 0-15, 1: lanes 16-31

    // Stage this k's phase-2 operand G into LDS (block cooperative).
    const float* Gk = G + (size_t)k * GMM_C * 16;
    for (int i = threadIdx.x; i < GMM_C * 16; i += 128) Gs[i] = Gk[i];
    __syncthreads();

    const int b0   = blockIdx.x * 64 + wave * 16;   // first b-row of this wave
    const int brow = b0 + n;
    const float x0 = kps[((size_t)brow * GMM_K + k) * 2 + 0];
    const float x1 = kps[((size_t)brow * GMM_K + k) * 2 + 1];

    // A operand (features of x), 16x8 split into two 16x4 K-steps.
    v2f A0, A1;
    if (half == 0) { A0.x = x0 * x0; A0.y = x0 * x1; A1.x = x1;   A1.y = 1.0f; }
    else           { A0.x = x1 * x1; A0.y = x0;      A1.x = 0.0f; A1.y = 0.0f; }

    const float* Mk = M + (size_t)k * 8 * GMM_C;
    float* cw = compS[wave];

    // ---- phase 1: logcomp = f(x) @ M, then comp = exp(logcomp), into LDS ----
#pragma unroll
    for (int t = 0; t < 8; ++t) {                   // 8 c-tiles of 16 columns
        const int c0 = t * 16 + n;
        v2f B0, B1;
        B0.x = Mk[(0 + 2 * half) * GMM_C + c0];
        B0.y = Mk[(1 + 2 * half) * GMM_C + c0];
        B1.x = Mk[(4 + 2 * half) * GMM_C + c0];
        B1.y = Mk[(5 + 2 * half) * GMM_C + c0];
        v8f acc = {};
        acc = __builtin_amdgcn_wmma_f32_16x16x4_f32(false, A0, false, B0,
                                                    (short)0, acc, false, false);
        acc = __builtin_amdgcn_wmma_f32_16x16x4_f32(false, A1, false, B1,
                                                    (short)0, acc, false, false);
#pragma unroll
        for (int v = 0; v < 8; ++v) {
            const int row = v + 8 * half;           // b-row within tile
            cw[row * GMM_C + t * 16 + n] = __expf(acc[v]);
        }
    }

    // ---- phase 2: S = comp @ G  (K = 128 contraction, 32 WMMA steps) ----
    v8f acc2 = {};
#pragma unroll
    for (int cc = 0; cc < GMM_C; cc += 4) {
        const int r = cc + 2 * half;
        v2f A2, B2;
        A2.x = cw[n * GMM_C + r];                   // comp[b=n][r], contiguous
        A2.y = cw[n * GMM_C + r + 1];               //   pair -> ds_load_b64
        B2.x = Gs[r * 16 + n];
        B2.y = Gs[(r + 1) * 16 + n];
        acc2 = __builtin_amdgcn_wmma_f32_16x16x4_f32(false, A2, false, B2,
                                                     (short)0, acc2, false, false);
    }

    // Spill the 16x16 S tile so each row's 7 sums land in one lane.
    float* sw = Ssh[wave];
#pragma unroll
    for (int v = 0; v < 8; ++v) sw[(v + 8 * half) * 16 + n] = acc2[v];

    // ---- epilogue: lanes 0-15 finish one b-row each (after all WMMAs) ----
    if (half == 0) {
        const float S0  = sw[n * 16 + 0];
        const float S1  = sw[n * 16 + 1];
        const float S2  = sw[n * 16 + 2];
        const float S3  = sw[n * 16 + 3];
        const float S4  = sw[n * 16 + 4];
        const float S5  = sw[n * 16 + 5];
        const float pdf = sw[n * 16 + 6];

        const float g0 = S4 - S0 * x0 - S1 * x1;    // -sum_c comp * P(x-mu)
        const float g1 = S5 - S2 * x0 - S3 * x1;
        float nrm = sqrtf(g0 * g0 + g1 * g1);
        nrm = fmaxf(nrm, 1e-30f);
        const float scale = __expf((kEps - nrm) * kInvLmb) / nrm;

        float* dens = out + (size_t)brow * 64 + 2 * k;
        dens[0] = pdf;
        dens[1] = pdf;
        float* gv = out + (size_t)Btot * 64 + (size_t)brow * 64 + 2 * k;
        gv[0] = g0 * scale;
        gv[1] = g1 * scale;
    }
}

// ---------------------------------------------------------------------------
// Harness entry point. Inputs: kps [B,32,2], weights [32,128],
// means [32,128,2], covariances [32,128,2,2]. Output: densities [B,64]
// followed by grad_vectors [B,64], concatenated flat.
// ---------------------------------------------------------------------------
extern "C" void kernel_launch(void* const* d_in, const int* in_sizes, int n_in,
                              void* d_out, int out_size, void* d_ws, size_t ws_size,
                              hipStream_t stream)
{
    const float* kps = (const float*)d_in[0];
    const float* w   = (const float*)d_in[1];
    const float* mu  = (const float*)d_in[2];
    const float* cov = (const float*)d_in[3];
    float* out = (float*)d_out;

    const int Btot = in_sizes[0] / (GMM_K * 2);     // 8192

    // Workspace: M [K][8][C] (128 KB) then G [K][C][16] (256 KB).
    float* Mws = (float*)d_ws;
    float* Gws = Mws + (size_t)GMM_K * 8 * GMM_C;

    gmm_precompute<<<(GMM_K * GMM_C + 255) / 256, 256, 0, stream>>>(w, mu, cov, Mws, Gws);

    dim3 grid(Btot / 64, GMM_K);
    gmm_wmma<<<grid, 128, 0, stream>>>(kps, Mws, Gws, out, Btot);
}